// FramesPositionalEncoding_9947144257847
// MI455X (gfx1250) — compile-verified
//
#include <hip/hip_runtime.h>

// Problem constants (from the reference)
#define T_LEN  4096
#define BATCH  32
#define CH     300
#define CH4    (CH / 4)      // 75 float4 per (t,b) row
#define WLEN   128
#define PE_MAX 5000

// ---------------------------------------------------------------------------
// Kernel A: per-batch segmentation.
// One block per batch (b = blockIdx.x), 128 threads (= W).
// 1) inclusive scan of durations -> ends; starts = ends - dur; total = ends[127]
// 2) for each frame t: seg_start(t) = largest start <= t (t < T-1),
//    or max start (t == T-1); pos = clip(t - seg_start, 0, PE_MAX-1);
//    write pos (or -1 if t >= total) to posOut[t*BATCH + b].
// ---------------------------------------------------------------------------
__global__ void seg_pos_kernel(const int* __restrict__ dur,
                               int* __restrict__ posOut) {
    const int b   = blockIdx.x;
    const int tid = threadIdx.x;

    __shared__ int s_scan[WLEN];
    __shared__ int s_start[WLEN];

    const int d = dur[b * WLEN + tid];
    s_scan[tid] = d;
    __syncthreads();

    // Hillis-Steele inclusive scan over 128 elements
    #pragma unroll
    for (int off = 1; off < WLEN; off <<= 1) {
        int v = (tid >= off) ? s_scan[tid - off] : 0;
        __syncthreads();
        s_scan[tid] += v;
        __syncthreads();
    }

    s_start[tid] = s_scan[tid] - d;   // starts (non-decreasing, starts[0] == 0)
    __syncthreads();

    const int total    = s_scan[WLEN - 1];
    const int maxStart = s_start[WLEN - 1];

    for (int t = tid; t < T_LEN; t += WLEN) {
        int seg;
        if (t == T_LEN - 1) {
            // all clipped scatter indices <= T-1, so max over all starts
            seg = maxStart;
        } else {
            // binary search: largest j with s_start[j] <= t (j=0 always valid)
            int lo = 0, hi = WLEN - 1;
            #pragma unroll 1
            while (lo < hi) {
                int mid = (lo + hi + 1) >> 1;
                if (s_start[mid] <= t) lo = mid; else hi = mid - 1;
            }
            seg = s_start[lo];
        }
        int pos = t - seg;
        pos = (pos < 0) ? 0 : ((pos > PE_MAX - 1) ? (PE_MAX - 1) : pos);
        posOut[t * BATCH + b] = (t < total) ? pos : -1;
    }
}

// ---------------------------------------------------------------------------
// Kernel B: out[t,b,:] = x[t,b,:] + (pos >= 0 ? pe[pos,:] : 0)
// Fully vectorized b128 stream: one float4 per thread.
// Total float4 count: 4096*32*75 = 9,830,400 (exactly 38400 blocks of 256).
// pe (6 MB) is L2-resident; gathered row is shared by 75 consecutive threads.
// ---------------------------------------------------------------------------
__global__ void add_pe_kernel(const float4* __restrict__ x4,
                              const float*  __restrict__ pe,
                              const int*    __restrict__ pos,
                              float4*       __restrict__ out4) {
    const unsigned idx = blockIdx.x * blockDim.x + threadIdx.x;
    const unsigned N4  = (unsigned)T_LEN * BATCH * CH4;
    if (idx >= N4) return;

    const unsigned r = idx / CH4;          // r = t*BATCH + b
    const unsigned q = idx - r * CH4;      // float4 column within the row

    const int p = pos[r];
    float4 v = x4[idx];                    // global_load_b128

    if (p >= 0) {
        const float* perow = pe + (unsigned)p * CH;
        __builtin_prefetch(perow, 0, 1);   // gfx1250: global_prefetch_b8
        const float4* perow4 = (const float4*)perow;  // 1200B row offset -> 16B aligned
        const float4 pv = perow4[q];       // global_load_b128 (L2 hit)
        v.x += pv.x; v.y += pv.y; v.z += pv.z; v.w += pv.w;
    }
    out4[idx] = v;                         // global_store_b128
}

// ---------------------------------------------------------------------------
// Host-side launcher
// Inputs (setup_inputs order): d_in[0]=x f32 (T,B,C), d_in[1]=pe f32 (5000,C),
// d_in[2]=text_duration i32 (B,W), d_in[3]=train (unused).
// Output: f32 (T,B,C). Workspace: pos table, BATCH*T_LEN ints = 512 KB.
// ---------------------------------------------------------------------------
extern "C" void kernel_launch(void* const* d_in, const int* in_sizes, int n_in,
                              void* d_out, int out_size, void* d_ws, size_t ws_size,
                              hipStream_t stream) {
    const float* x   = (const float*)d_in[0];
    const float* pe  = (const float*)d_in[1];
    const int*   dur = (const int*)d_in[2];
    (void)in_sizes; (void)n_in; (void)out_size; (void)ws_size;

    int* posTable = (int*)d_ws;   // T_LEN*BATCH ints = 512 KB

    seg_pos_kernel<<<BATCH, WLEN, 0, stream>>>(dur, posTable);

    const unsigned N4 = (unsigned)T_LEN * BATCH * CH4;  // 9,830,400
    const unsigned threads = 256;
    const unsigned blocks  = (N4 + threads - 1) / threads;  // 38,400
    add_pe_kernel<<<blocks, threads, 0, stream>>>((const float4*)x, pe, posTable,
                                                  (float4*)d_out);
}